// ImportancePooling_28424093564958
// MI455X (gfx1250) — compile-verified
//
#include <hip/hip_runtime.h>

// CDNA5 / gfx1250: wave32, WMMA f32 16x16x4 (full f32 precision, matches the
// f32 reference). Kernel is L2-bandwidth bound (x fits in 192MB L2; ~819MB of
// gather traffic, ~77MB HBM floor ~= 3.3us), so the broadcast-A WMMA trick's
// 1/16 matrix utilization is free and keeps the reduction on the matrix path.

typedef __attribute__((ext_vector_type(2))) float v2f;
typedef __attribute__((ext_vector_type(8))) float v8f;

#define KNBR 32
#define DFEAT 64
#define WAVES_PER_BLOCK 8

__global__ __launch_bounds__(256) void ImportancePooling_28424093564958_kernel(
    const float* __restrict__ x,        // [N, 64]
    const int*   __restrict__ nbr,      // [N, 32]
    const float* __restrict__ wgt,      // [N, 32]
    float*       __restrict__ out,      // [N, 64]
    int n_nodes)
{
    const int lane = threadIdx.x & 31;
    const int wave = threadIdx.x >> 5;
    const int node = blockIdx.x * WAVES_PER_BLOCK + wave;
    if (node >= n_nodes) return;            // wave-uniform: EXEC stays all-1s

    // Lane k owns neighbor k of this node (K == wave width == 32).
    const float w_lane  = wgt[(size_t)node * KNBR + lane];
    const int   id_lane = nbr[(size_t)node * KNBR + lane];
    const int   rowbase_lane = id_lane * DFEAT;   // row start in floats

    // wsum = wave-wide sum of weights; normalize (reference: keep raw if <= 0)
    float ws = w_lane;
    #pragma unroll
    for (int off = 16; off > 0; off >>= 1)
        ws += __shfl_xor(ws, off, 32);
    const float wn_lane = (ws > 0.0f) ? (w_lane / ws) : w_lane;

    // WMMA f32 16x16x4 register layout (ISA 7.12.2):
    //   A(16x4): VGPR v, lane-half h  -> K = v + 2*h   (M = lane & 15)
    //   B(4x16): same stride-=-half pattern assumed -> row = v + 2*h, col = lane & 15
    //   C(16x16): VGPR0 holds rows {0,8}; with A rows replicated every C entry
    //             in VGPR0 equals out[chunk*16 + (lane & 15)].
    const int hi  = (lane >> 4) & 1;     // which lane-half
    const int col = lane & 15;           // B column / output feature-in-chunk

    v8f acc0 = {}, acc1 = {}, acc2 = {}, acc3 = {};

    #pragma unroll
    for (int c = 0; c < 8; ++c) {
        const int k0 = 4 * c;
        // A: broadcast normalized weights into every row (same placement fn as B)
        v2f a;
        a.x = __shfl(wn_lane, k0 + 2 * hi,     32);   // slot (v=0, h=hi) -> k0+2h
        a.y = __shfl(wn_lane, k0 + 2 * hi + 1, 32);   // slot (v=1, h=hi) -> k0+2h+1

        // Row bases for the two B registers this lane contributes to
        const int rb0 = __shfl(rowbase_lane, k0 + 2 * hi,     32);
        const int rb1 = __shfl(rowbase_lane, k0 + 2 * hi + 1, 32);

        #pragma unroll
        for (int j4 = 0; j4 < 4; ++j4) {
            // Gather directly into the B layout: each lane-half reads 64
            // contiguous bytes of one neighbor row (cache-line friendly).
            v2f b;
            b.x = x[(size_t)rb0 + j4 * 16 + col];
            b.y = x[(size_t)rb1 + j4 * 16 + col];

            v8f* accp = (j4 == 0) ? &acc0 : (j4 == 1) ? &acc1
                       : (j4 == 2) ? &acc2 : &acc3;
            *accp = __builtin_amdgcn_wmma_f32_16x16x4_f32(
                        /*neg_a=*/false, a, /*neg_b=*/false, b,
                        /*c_mod=*/(short)0, *accp,
                        /*reuse_a=*/false, /*reuse_b=*/false);
        }
    }

    // Rows of C are replicated, so VGPR0 (element 0) of each accumulator holds
    // feature chunk j4 at column (lane & 15) in BOTH lane halves. Pack two
    // chunks per full-wave coalesced store (2 x 128B per node).
    float* optr = out + (size_t)node * DFEAT;
    const float v0 = (lane < 16) ? acc0[0] : acc1[0];   // feats 0..31
    const float v1 = (lane < 16) ? acc2[0] : acc3[0];   // feats 32..63
    optr[lane]      = v0;
    optr[32 + lane] = v1;
}

extern "C" void kernel_launch(void* const* d_in, const int* in_sizes, int n_in,
                              void* d_out, int out_size, void* d_ws, size_t ws_size,
                              hipStream_t stream) {
    const float* x   = (const float*)d_in[0];   // [N, 64] f32
    const int*   nbr = (const int*)  d_in[1];   // [N, 32] i32
    const float* wgt = (const float*)d_in[2];   // [N, 32] f32
    float*       out = (float*)d_out;           // [N, 64] f32

    const int n = in_sizes[0] / DFEAT;          // N = 100000
    const int grid = (n + WAVES_PER_BLOCK - 1) / WAVES_PER_BLOCK;

    hipLaunchKernelGGL(ImportancePooling_28424093564958_kernel,
                       dim3(grid), dim3(256), 0, stream,
                       x, nbr, wgt, out, n);
}